// UpFIRDn2dJIT_26525718020493
// MI455X (gfx1250) — compile-verified
//
#include <hip/hip_runtime.h>

typedef __attribute__((ext_vector_type(2))) float v2f;
typedef __attribute__((ext_vector_type(8))) float v8f;

#define GAIN 1.41421356237309515f
#define SLOPE 0.2f

__device__ __forceinline__ v8f wmma4(v2f a, v2f b, v8f c) {
  // D = A(16x4 f32) * B(4x16 f32) + C(16x16 f32)
  return __builtin_amdgcn_wmma_f32_16x16x4_f32(false, a, false, b, (short)0, c,
                                               false, false);
}

__device__ __forceinline__ int clampi(int v, int lo, int hi) {
  return v < lo ? lo : (v > hi ? hi : v);
}

// Combined (nearest-upsample ∘ 12-tap FIR, pad 5/6) Toeplitz value:
// contribution of source row/col (base+j) to upsampled-filtered position u'
// (tile-local, tile base even, base = U0/2 - 3).
__device__ __forceinline__ float tu_value(const float* __restrict__ f, int j, int up) {
  int d = j - (up >> 1);
  float v = 0.0f;
  if ((up & 1) == 0) {               // even phase: 7 taps g0[d] = f[2d-1]+f[2d]
    if (d >= 0 && d <= 6) {
      int k = 2 * d;
      float lo = (k >= 1)  ? f[k - 1] : 0.0f;
      float hi = (k <= 11) ? f[k]     : 0.0f;
      v = lo + hi;
    }
  } else {                           // odd phase: 6 taps g1[t] = f[2t]+f[2t+1]
    int t = d - 1;
    if (t >= 0 && t <= 5) v = f[2 * t] + f[2 * t + 1];
  }
  return v;
}

// Down-conv (12-tap FIR, pad 5/6, then ::2) Toeplitz: TD[j][m'] = f[j - 2m'].
__device__ __forceinline__ float td_value(const float* __restrict__ f, int j, int mp) {
  int k = j - 2 * mp;
  return (k >= 0 && k < 12) ? f[k] : 0.0f;
}

// ---------------- Pass 1: up-H (x[512,130,130] crop+bias -> t1[512,256,128])
// out(16u' x 16n) = TU(16u' x 16j) * X(16j x 16n); 8 col tiles (full row) per wave.
__global__ __launch_bounds__(256) void pass1_up_h(
    const float* __restrict__ x, const float* __restrict__ bias,
    const float* __restrict__ fup, float* __restrict__ t1) {
  __shared__ float sTU[16][16];  // [u'][j]
  for (int i = threadIdx.x; i < 256; i += 256)
    sTU[i >> 4][i & 15] = tu_value(fup, i & 15, i >> 4);
  __syncthreads();

  const int lane = threadIdx.x & 31;
  const int half = lane >> 4;
  const int mrow = lane & 15;

  int t = blockIdx.x * 8 + (threadIdx.x >> 5);
  const int utile = t & 15; t >>= 4;   // 16 row tiles (height 256)
  const int ch = t;                    // 512 planes

  const int rbase = 8 * utile - 3;
  const float bb  = bias[ch & 63];
  // crop offsets (+1 row, +1 col) folded into the base pointer
  const float* xb = x + (size_t)ch * 130 * 130 + 131 + mrow;
  float* ob = t1 + ((size_t)ch * 256 + 16 * utile + 8 * half) * 128 + mrow;

  v8f acc[8] = {};
#pragma unroll
  for (int c = 0; c < 4; ++c) {
    const int j0 = 4 * c + 2 * half;
    v2f a;  // A[u'][j] = TU(j,u'); zero columns whose source row is OOB
    a.x = sTU[mrow][j0];
    a.y = sTU[mrow][j0 + 1];
    const int r0 = rbase + j0, r1 = r0 + 1;
    if ((unsigned)r0 >= 128u) a.x = 0.0f;
    if ((unsigned)r1 >= 128u) a.y = 0.0f;
    const float* p0 = xb + (size_t)clampi(r0, 0, 127) * 130;
    const float* p1 = xb + (size_t)clampi(r1, 0, 127) * 130;
#pragma unroll
    for (int nt = 0; nt < 8; ++nt) {
      v2f b;  // B[j][n] = x[r][col] + bias (unconditional, clamped addr)
      b.x = p0[16 * nt] + bb;
      b.y = p1[16 * nt] + bb;
      acc[nt] = wmma4(a, b, acc[nt]);
    }
  }
#pragma unroll
  for (int g = 0; g < 8; ++g)
#pragma unroll
    for (int nt = 0; nt < 8; ++nt)
      ob[(size_t)g * 128 + 16 * nt] = acc[nt][g];
}

// ---------------- Pass 2: up-W + leaky (t1[512,256,128] -> t2[512,256,256])
// out(16m x 16v') = T1(16m x 16j) * TU(16j x 16v'); 8 row tiles per wave.
__global__ __launch_bounds__(256) void pass2_up_w_lrelu(
    const float* __restrict__ t1, const float* __restrict__ fup,
    float* __restrict__ t2) {
  __shared__ float sTU[16][16];  // [v'][j]
  for (int i = threadIdx.x; i < 256; i += 256)
    sTU[i >> 4][i & 15] = tu_value(fup, i & 15, i >> 4);
  __syncthreads();

  const int lane = threadIdx.x & 31;
  const int half = lane >> 4;
  const int mrow = lane & 15;

  int t = blockIdx.x * 8 + (threadIdx.x >> 5);
  const int vtile  = t & 15; t >>= 4;  // 16 col tiles (width 256)
  const int mgroup = t & 1;  t >>= 1;  // 2 groups of 8 row tiles
  const int ch = t;

  const int cbase = 8 * vtile - 3;
  const float* ab = t1 + ((size_t)ch * 256 + 128 * mgroup + mrow) * 128;
  float* ob = t2 + ((size_t)ch * 256 + 128 * mgroup + 8 * half) * 256
            + 16 * vtile + mrow;

  v8f acc[8] = {};
#pragma unroll
  for (int c = 0; c < 4; ++c) {
    const int j0 = 4 * c + 2 * half;
    const int c0 = cbase + j0, c1 = c0 + 1;
    v2f b;  // B[j][v'] = TU(j,v'); zero rows whose source col is OOB
    b.x = ((unsigned)c0 < 128u) ? sTU[mrow][j0]     : 0.0f;
    b.y = ((unsigned)c1 < 128u) ? sTU[mrow][j0 + 1] : 0.0f;
    const float* q0 = ab + clampi(c0, 0, 127);
    const float* q1 = ab + clampi(c1, 0, 127);
#pragma unroll
    for (int nt = 0; nt < 8; ++nt) {
      v2f a;  // A[m][j] = t1[row][col] (unconditional, clamped addr)
      a.x = q0[2048 * nt];   // 16 rows * 128
      a.y = q1[2048 * nt];
      acc[nt] = wmma4(a, b, acc[nt]);
    }
  }
#pragma unroll
  for (int g = 0; g < 8; ++g)
#pragma unroll
    for (int nt = 0; nt < 8; ++nt) {
      float v = acc[nt][g];
      v = (v >= 0.0f ? v : SLOPE * v) * GAIN;
      ob[(size_t)(16 * nt + g) * 256] = v;
    }
}

// ---------------- Pass 3: dn-H + ::2 rows (t2[512,256,256] -> t3[512,128,256])
// out(16m' x 16n) = TD(16m' x 44j) * Y(44j x 16n); 8 col tiles per wave.
__global__ __launch_bounds__(256) void pass3_dn_h(
    const float* __restrict__ t2, const float* __restrict__ fdn,
    float* __restrict__ t3) {
  __shared__ float sTD[16][48];  // [m'][j], j<44 used
  for (int i = threadIdx.x; i < 16 * 48; i += 256) {
    int mp = i / 48, j = i % 48;
    sTD[mp][j] = (j < 44) ? td_value(fdn, j, mp) : 0.0f;
  }
  __syncthreads();

  const int lane = threadIdx.x & 31;
  const int half = lane >> 4;
  const int mrow = lane & 15;

  int t = blockIdx.x * 8 + (threadIdx.x >> 5);
  const int ngroup = t & 1; t >>= 1;   // 2 groups of 8 col tiles (width 256)
  const int mtile  = t & 7; t >>= 3;   // 8 row tiles (out height 128)
  const int ch = t;

  const int rbase = 32 * mtile - 5;
  const float* xb = t2 + (size_t)ch * 256 * 256 + 128 * ngroup + mrow;
  float* ob = t3 + ((size_t)ch * 128 + 16 * mtile + 8 * half) * 256
            + 128 * ngroup + mrow;

  v8f acc[8] = {};
#pragma unroll
  for (int c = 0; c < 11; ++c) {
    const int j0 = 4 * c + 2 * half;
    v2f a;  // A[m'][j] = TD[j][m']; zero columns whose source row is OOB
    a.x = sTD[mrow][j0];
    a.y = sTD[mrow][j0 + 1];
    const int r0 = rbase + j0, r1 = r0 + 1;
    if ((unsigned)r0 >= 256u) a.x = 0.0f;
    if ((unsigned)r1 >= 256u) a.y = 0.0f;
    const float* p0 = xb + (size_t)clampi(r0, 0, 255) * 256;
    const float* p1 = xb + (size_t)clampi(r1, 0, 255) * 256;
#pragma unroll
    for (int nt = 0; nt < 8; ++nt) {
      v2f b;
      b.x = p0[16 * nt];
      b.y = p1[16 * nt];
      acc[nt] = wmma4(a, b, acc[nt]);
    }
  }
#pragma unroll
  for (int g = 0; g < 8; ++g)
#pragma unroll
    for (int nt = 0; nt < 8; ++nt)
      ob[(size_t)g * 256 + 16 * nt] = acc[nt][g];
}

// ---------------- Pass 4: dn-W + ::2 cols (t3[512,128,256] -> out[512,128,128])
// out(16m x 16n') = T3(16m x 44j) * TD(44j x 16n'); 8 row tiles (full col) per wave.
__global__ __launch_bounds__(256) void pass4_dn_w(
    const float* __restrict__ t3, const float* __restrict__ fdn,
    float* __restrict__ out) {
  __shared__ float sTD[16][48];
  for (int i = threadIdx.x; i < 16 * 48; i += 256) {
    int mp = i / 48, j = i % 48;
    sTD[mp][j] = (j < 44) ? td_value(fdn, j, mp) : 0.0f;
  }
  __syncthreads();

  const int lane = threadIdx.x & 31;
  const int half = lane >> 4;
  const int mrow = lane & 15;

  int t = blockIdx.x * 8 + (threadIdx.x >> 5);
  const int ntile = t & 7; t >>= 3;    // 8 col tiles (out width 128)
  const int ch = t;

  const int cbase = 32 * ntile - 5;
  const float* ab = t3 + ((size_t)ch * 128 + mrow) * 256;
  float* ob = out + ((size_t)ch * 128 + 8 * half) * 128 + 16 * ntile + mrow;

  v8f acc[8] = {};
#pragma unroll
  for (int c = 0; c < 11; ++c) {
    const int j0 = 4 * c + 2 * half;
    const int c0 = cbase + j0, c1 = c0 + 1;
    v2f b;  // B[j][n'] = TD[j][n']; zero rows whose source col is OOB
    b.x = ((unsigned)c0 < 256u) ? sTD[mrow][j0]     : 0.0f;
    b.y = ((unsigned)c1 < 256u) ? sTD[mrow][j0 + 1] : 0.0f;
    const float* q0 = ab + clampi(c0, 0, 255);
    const float* q1 = ab + clampi(c1, 0, 255);
#pragma unroll
    for (int nt = 0; nt < 8; ++nt) {
      v2f a;
      a.x = q0[4096 * nt];   // 16 rows * 256
      a.y = q1[4096 * nt];
      acc[nt] = wmma4(a, b, acc[nt]);
    }
  }
#pragma unroll
  for (int g = 0; g < 8; ++g)
#pragma unroll
    for (int nt = 0; nt < 8; ++nt)
      ob[(size_t)(16 * nt + g) * 128] = acc[nt][g];
}

extern "C" void kernel_launch(void* const* d_in, const int* in_sizes, int n_in,
                              void* d_out, int out_size, void* d_ws, size_t ws_size,
                              hipStream_t stream) {
  (void)in_sizes; (void)n_in; (void)out_size; (void)ws_size;
  const float* x    = (const float*)d_in[0];
  const float* bias = (const float*)d_in[1];
  const float* fup  = (const float*)d_in[2];
  const float* fdn  = (const float*)d_in[3];
  float* out = (float*)d_out;

  // Workspace: t1 (64 MB) | t2 (128 MB); t3 reuses t1's region after pass 2.
  float* t1 = (float*)d_ws;
  float* t2 = (float*)((char*)d_ws + (size_t)512 * 256 * 128 * 4);
  float* t3 = (float*)d_ws;

  dim3 blk(256);
  pass1_up_h      <<<512 * 16     / 8, blk, 0, stream>>>(x, bias, fup, t1);
  pass2_up_w_lrelu<<<512 * 2 * 16 / 8, blk, 0, stream>>>(t1, fup, t2);
  pass3_dn_h      <<<512 * 8 * 2  / 8, blk, 0, stream>>>(t2, fdn, t3);
  pass4_dn_w      <<<512 * 8      / 8, blk, 0, stream>>>(t3, fdn, out);
}